// UnifiedFlowFrag_88776974008839
// MI455X (gfx1250) — compile-verified
//
#include <hip/hip_runtime.h>
#include <stdint.h>

#define NF_EPS 1e-4f
#define REC    16      // floats per fragment accumulator record (64B)
#define TILE   1024    // atoms per block tile
#define TPB    256

__device__ __forceinline__ void atomic_add_f32(float* p, float v) {
  __hip_atomic_fetch_add(p, v, __ATOMIC_RELAXED, __HIP_MEMORY_SCOPE_AGENT);
}

// ---------------------------------------------------------------- zero ws
__global__ void nf_zero(float4* __restrict__ ws, int n4) {
  int i = blockIdx.x * blockDim.x + threadIdx.x;
  if (i < n4) ws[i] = make_float4(0.f, 0.f, 0.f, 0.f);
}

// ------------------------------------------------------- per-atom scatter
__device__ __forceinline__ void nf_accum(float* __restrict__ ws,
                                         const float* __restrict__ T_frag,
                                         int id,
                                         float fx, float fy, float fz,
                                         float px, float py, float pz) {
  float tx = T_frag[id * 3 + 0];
  float ty = T_frag[id * 3 + 1];
  float tz = T_frag[id * 3 + 2];
  float rx = px - tx, ry = py - ty, rz = pz - tz;
  // torque = cross(r, f)
  float qx = ry * fz - rz * fy;
  float qy = rz * fx - rx * fz;
  float qz = rx * fy - ry * fx;
  float* rec = ws + (size_t)id * REC;
  atomic_add_f32(rec + 0, fx);
  atomic_add_f32(rec + 1, fy);
  atomic_add_f32(rec + 2, fz);
  atomic_add_f32(rec + 3, 1.0f);
  atomic_add_f32(rec + 4, qx);
  atomic_add_f32(rec + 5, qy);
  atomic_add_f32(rec + 6, qz);
  atomic_add_f32(rec + 7, rx * rx);
  atomic_add_f32(rec + 8, ry * ry);
  atomic_add_f32(rec + 9, rz * rz);
  atomic_add_f32(rec + 10, rx * ry);
  atomic_add_f32(rec + 11, rx * rz);
  atomic_add_f32(rec + 12, ry * rz);
}

__global__ __launch_bounds__(TPB) void nf_atoms(
    const float* __restrict__ f_atom, const float* __restrict__ atom_pos,
    const float* __restrict__ T_frag, const int* __restrict__ frag_id,
    float* __restrict__ ws, int n_atom, int full_tiles) {
  // [0, TILE*3): forces, [TILE*3, TILE*6): positions  (24 KB)
  __shared__ __align__(16) float smem[TILE * 3 * 2];
  const int t    = threadIdx.x;
  const int tile = blockIdx.x;

  if (tile < full_tiles) {
    // LDS byte offset of smem within the wave's allocation: for an AS(3)
    // object the generic pointer's low 32 bits are the LDS offset. The
    // ptrtoint also escapes smem so the asm "memory" clobber is honored
    // and the later LDS reads are not folded away.
    const unsigned lds_base =
        (unsigned)(unsigned long long)(uintptr_t)(void*)smem;

    // --- CDNA5 async bulk copy: tile's f/pos streams -> LDS ---------------
    const unsigned long long fbase =
        (unsigned long long)f_atom + (unsigned long long)tile * (TILE * 12);
    const unsigned long long pbase =
        (unsigned long long)atom_pos + (unsigned long long)tile * (TILE * 12);
#pragma unroll
    for (int r = 0; r < 3; ++r) {
      unsigned voff = (unsigned)(t * 16 + r * 4096);   // 16B per lane per round
      unsigned ldsf = lds_base + voff;
      unsigned ldsp = lds_base + (unsigned)(TILE * 12) + voff;
      asm volatile("global_load_async_to_lds_b128 %0, %1, %2"
                   :: "v"(ldsf), "v"(voff), "s"(fbase) : "memory");
      asm volatile("global_load_async_to_lds_b128 %0, %1, %2"
                   :: "v"(ldsp), "v"(voff), "s"(pbase) : "memory");
    }
    asm volatile("s_wait_asynccnt 0x0" ::: "memory");
    __syncthreads();

    const int base = tile * TILE;
    int4 fid = ((const int4*)(frag_id + base))[t];     // 4 atoms per thread
#pragma unroll
    for (int k = 0; k < 4; ++k) {
      int la = t * 4 + k;
      float fx = smem[la * 3 + 0];
      float fy = smem[la * 3 + 1];
      float fz = smem[la * 3 + 2];
      float px = smem[TILE * 3 + la * 3 + 0];
      float py = smem[TILE * 3 + la * 3 + 1];
      float pz = smem[TILE * 3 + la * 3 + 2];
      int id = (k == 0) ? fid.x : (k == 1) ? fid.y : (k == 2) ? fid.z : fid.w;
      nf_accum(ws, T_frag, id, fx, fy, fz, px, py, pz);
    }
  } else {
    // --- tail: leftover atoms, plain loads --------------------------------
    for (int a = full_tiles * TILE + t; a < n_atom; a += TPB) {
      float fx = f_atom[a * 3 + 0], fy = f_atom[a * 3 + 1], fz = f_atom[a * 3 + 2];
      float px = atom_pos[a * 3 + 0], py = atom_pos[a * 3 + 1], pz = atom_pos[a * 3 + 2];
      nf_accum(ws, T_frag, frag_id[a], fx, fy, fz, px, py, pz);
    }
  }
}

// ----------------------------------------------------- per-fragment solve
__global__ void nf_finalize(const float* __restrict__ ws,
                            const int* __restrict__ frag_sizes,
                            float* __restrict__ out_v, float* __restrict__ out_w,
                            int n_frag) {
  int i = blockIdx.x * blockDim.x + threadIdx.x;
  if (i >= n_frag) return;
  const float4* r4 = (const float4*)(ws + (size_t)i * REC);
  float4 A = r4[0];  // Fx Fy Fz count
  float4 B = r4[1];  // Tx Ty Tz Sxx
  float4 C = r4[2];  // Syy Szz Sxy Sxz
  float4 D = r4[3];  // Syz pad pad pad

  float inv = 1.0f / fmaxf(A.w, 1.0f);
  float vx = A.x * inv, vy = A.y * inv, vz = A.z * inv;

  float xx = B.w, yy = C.x, zz = C.y, xy = C.z, xz = C.w, yz = D.x;
  // I = tr(S)*E - S + eps*E  (symmetric positive definite)
  float a  = yy + zz + NF_EPS;
  float dd = xx + zz + NF_EPS;
  float ff = xx + yy + NF_EPS;
  float b = -xy, c = -xz, e = -yz;

  float A00 = dd * ff - e * e;
  float A01 = c * e - b * ff;
  float A02 = b * e - c * dd;
  float det = a * A00 + b * A01 + c * A02;
  float A11 = a * ff - c * c;
  float A12 = b * c - a * e;
  float A22 = a * dd - b * b;
  float idet = 1.0f / det;

  float t0 = B.x, t1 = B.y, t2 = B.z;
  float wx = (A00 * t0 + A01 * t1 + A02 * t2) * idet;
  float wy = (A01 * t0 + A11 * t1 + A12 * t2) * idet;
  float wz = (A02 * t0 + A12 * t1 + A22 * t2) * idet;

  if (frag_sizes[i] <= 1) { wx = 0.f; wy = 0.f; wz = 0.f; }

  out_v[i * 3 + 0] = vx;
  out_v[i * 3 + 1] = vy;
  out_v[i * 3 + 2] = vz;
  out_w[i * 3 + 0] = wx;
  out_w[i * 3 + 1] = wy;
  out_w[i * 3 + 2] = wz;
}

// ------------------------------------------------------------------ launch
extern "C" void kernel_launch(void* const* d_in, const int* in_sizes, int n_in,
                              void* d_out, int out_size, void* d_ws, size_t ws_size,
                              hipStream_t stream) {
  const float* f_atom     = (const float*)d_in[0];
  const float* atom_pos   = (const float*)d_in[1];
  const float* T_frag     = (const float*)d_in[2];
  const int*   frag_id    = (const int*)d_in[3];
  const int*   frag_sizes = (const int*)d_in[5];

  int n_atom = in_sizes[0] / 3;
  int n_frag = in_sizes[5];

  float* ws    = (float*)d_ws;                 // n_frag * 16 floats = 12.8 MB
  float* out_v = (float*)d_out;
  float* out_w = out_v + (size_t)n_frag * 3;

  int n4 = (n_frag * REC) / 4;
  nf_zero<<<(n4 + 255) / 256, 256, 0, stream>>>((float4*)ws, n4);

  int full_tiles = n_atom / TILE;
  nf_atoms<<<full_tiles + 1, TPB, 0, stream>>>(f_atom, atom_pos, T_frag,
                                               frag_id, ws, n_atom, full_tiles);

  nf_finalize<<<(n_frag + 255) / 256, 256, 0, stream>>>(ws, frag_sizes,
                                                        out_v, out_w, n_frag);
}